// DeepCAD_1958505087412
// MI455X (gfx1250) — compile-verified
//
#include <hip/hip_runtime.h>
#include <hip/hip_bf16.h>

typedef __attribute__((ext_vector_type(2))) float v2f;
typedef __attribute__((ext_vector_type(8))) float v8f;
typedef __attribute__((ext_vector_type(4))) unsigned int u32x4;
typedef __attribute__((ext_vector_type(8))) int i32x8;
typedef __attribute__((ext_vector_type(4))) int i32x4;

__device__ __forceinline__ float sigm(float x) { return 1.0f / (1.0f + __expf(-x)); }

// ---------------------------------------------------------------------------
// Generic f32 WMMA GEMM:  C[z] = A[z] @ op(B[z]) + bias  (op = T if BT)
//   BT=true : B stored (N,K) row-major (PyTorch Linear weight)  -> C = A @ B^T
//   BT=false: B stored (K,N) row-major                          -> C = A @ B
// Each wave computes a 16(M) x 64(N) tile with V_WMMA_F32_16X16X4_F32.
// Requires M % 16 == 0, K % 4 == 0. N handled in 16-wide subtiles.
// Epilogue: + bias[col], * rowmask[z,row], optional ReLU.
// ---------------------------------------------------------------------------
template <bool BT, bool RELU>
__global__ __launch_bounds__(256)
void gemm_wmma(const float* __restrict__ A, int lda, int sA,
               const float* __restrict__ Bm, int ldb, int sB,
               const float* __restrict__ bias,
               float* __restrict__ C, int ldc, int sC,
               const float* __restrict__ rowmask, int mstride,
               int M, int N, int K)
{
  const int lane   = threadIdx.x & 31;
  const int wid    = threadIdx.x >> 5;
  const int tilesN = (N + 63) >> 6;
  const int tilesM = M >> 4;
  const int id     = blockIdx.x * (blockDim.x >> 5) + wid;
  if (id >= tilesM * tilesN) return;
  const int tm = id / tilesN, tn = id % tilesN;
  const int m0 = tm << 4, n0 = tn << 6;
  const int z  = blockIdx.z;
  A  += (long)z * sA;
  Bm += (long)z * sB;
  C  += (long)z * sC;
  const int half = lane >> 4;   // K sub-offset selector (A/B frag layout)
  const int r    = lane & 15;   // M row (A) / N col (B,C)

  const v8f vzero = {0,0,0,0,0,0,0,0};
  v8f acc0 = vzero, acc1 = vzero, acc2 = vzero, acc3 = vzero;

  const float* Arow = A + (m0 + r) * lda + 2 * half;
  const bool s1 = (n0 + 16) < N, s2 = (n0 + 32) < N, s3 = (n0 + 48) < N;

  for (int k = 0; k < K; k += 4) {
    v2f a = *(const v2f*)(Arow + k);
#define GEMM_SUB(S, ACC)                                                        \
    {                                                                           \
      const int nc = n0 + 16 * (S) + r;                                         \
      v2f b;                                                                    \
      if (BT) {                                                                 \
        b = *(const v2f*)(Bm + nc * ldb + k + 2 * half);                        \
      } else {                                                                  \
        b.x = Bm[(k + 2 * half) * ldb + nc];                                    \
        b.y = Bm[(k + 2 * half + 1) * ldb + nc];                                \
      }                                                                         \
      ACC = __builtin_amdgcn_wmma_f32_16x16x4_f32(false, a, false, b,           \
                                                  (short)0, ACC, false, false); \
    }
    GEMM_SUB(0, acc0)
    if (s1) GEMM_SUB(1, acc1)
    if (s2) GEMM_SUB(2, acc2)
    if (s3) GEMM_SUB(3, acc3)
#undef GEMM_SUB
  }

#define GEMM_EPI(S, ACC)                                                        \
  {                                                                             \
    const int col = n0 + 16 * (S) + r;                                          \
    const float bv = bias ? bias[col] : 0.0f;                                   \
    _Pragma("unroll")                                                           \
    for (int v = 0; v < 8; v++) {                                               \
      const int row = m0 + v + 8 * half;                                        \
      float val = ACC[v] + bv;                                                  \
      if (rowmask) val *= rowmask[z * mstride + row];                           \
      if (RELU) val = fmaxf(val, 0.0f);                                         \
      C[row * ldc + col] = val;                                                 \
    }                                                                           \
  }
  GEMM_EPI(0, acc0)
  if (s1) GEMM_EPI(1, acc1)
  if (s2) GEMM_EPI(2, acc2)
  if (s3) GEMM_EPI(3, acc3)
#undef GEMM_EPI
}

// ---------------------------------------------------------------------------
// Attention row/col score vectors: si = h @ aw[:256] + ab ; sj = h @ aw[256:]
// ---------------------------------------------------------------------------
__global__ __launch_bounds__(256)
void attn_vec(const float* __restrict__ h, const float* __restrict__ aw,
              const float* __restrict__ ab, float* __restrict__ si,
              float* __restrict__ sj)
{
  const int row = blockIdx.x * blockDim.x + threadIdx.x;  // 0..4095
  const float* hr = h + row * 256;
  float a = 0.0f, b = 0.0f;
  for (int k = 0; k < 256; k++) {
    const float x = hr[k];
    a += x * aw[k];
    b += x * aw[256 + k];
  }
  si[row] = a + ab[0];
  sj[row] = b;
}

// ---------------------------------------------------------------------------
// Attention weights: watt[b,i,:] = score / (sum(score) + 1e-8), one wave/row.
// score = sigmoid(si+sj) * adj * mask_i * mask_j
// ---------------------------------------------------------------------------
__global__ __launch_bounds__(256)
void attn_watt(const float* __restrict__ si, const float* __restrict__ sj,
               const float* __restrict__ adj, const float* __restrict__ mask,
               float* __restrict__ watt)
{
  const int gw   = (blockIdx.x * blockDim.x + threadIdx.x) >> 5;  // 0..4095
  const int lane = threadIdx.x & 31;
  const int b = gw >> 9, i = gw & 511;
  const float s_i = si[b * 512 + i];
  const float m_i = mask[b * 512 + i];
  const float* arow = adj + (b * 512 + i) * 512;
  float vals[16];
  float sum = 0.0f;
#pragma unroll
  for (int k = 0; k < 16; k++) {
    const int j = lane + 32 * k;
    float sc = sigm(s_i + sj[b * 512 + j]);
    sc *= arow[j] * m_i * mask[b * 512 + j];
    vals[k] = sc;
    sum += sc;
  }
  for (int off = 16; off > 0; off >>= 1) sum += __shfl_xor(sum, off, 32);
  const float inv = 1.0f / (sum + 1e-8f);
  float* wrow = watt + (b * 512 + i) * 512;
#pragma unroll
  for (int k = 0; k < 16; k++) wrow[lane + 32 * k] = vals[k] * inv;
}

// ---------------------------------------------------------------------------
// Tensor Data Mover: DMA an (8 rows x 1024 f32) pre-activation tile
// (row stride T*1024 elements) from global into LDS at byte offset lds_off.
// D# packing per CDNA5 ISA:
//   group0: count=1 | lds_addr | global_addr[56:0] | type=2
//   group1: data_size=2 (4B), tensor_dim0=1024, tensor_dim1=4096,
//           tile_dim0=1024, tile_dim1=8, tensor_dim0_stride=T*1024
//   groups 2/3: zero (2D tile)
// Tracked with TENSORcnt; consumed after s_wait_tensorcnt 0 + barrier.
// ---------------------------------------------------------------------------
#if __has_builtin(__builtin_amdgcn_tensor_load_to_lds) && \
    __has_builtin(__builtin_amdgcn_s_wait_tensorcnt)
#define HAVE_TDM 1
#else
#define HAVE_TDM 0
#endif

#if HAVE_TDM
__device__ __forceinline__ void tdm_load_pre_tile(const float* gaddr,
                                                  unsigned lds_off, int T)
{
  const unsigned long long ga = (unsigned long long)(uintptr_t)gaddr;
  u32x4 g0;
  g0.x = 1u;                                      // count=1 (user, load)
  g0.y = lds_off;                                 // LDS byte address
  g0.z = (unsigned)(ga & 0xFFFFFFFFu);            // global_addr[31:0]
  g0.w = (unsigned)((ga >> 32) & 0x01FFFFFFu)     // global_addr[56:32]
       | (2u << 30);                              // type=2 ("image")
  i32x8 g1;
  g1[0] = 0x00020000;                             // data_size=2 -> 4 bytes
  g1[1] = (int)(1024u << 16);                     // tensor_dim0[15:0]=1024
  g1[2] = (int)(4096u << 16);                     // td0[31:16]=0 | td1[15:0]=4096
  g1[3] = (int)(1024u << 16);                     // td1[31:16]=0 | tile_dim0=1024
  g1[4] = 8;                                      // tile_dim1=8, tile_dim2=0
  g1[5] = (int)((unsigned)T * 1024u);             // tensor_dim0_stride[31:0]
  g1[6] = 0;                                      // stride[47:32]=0
  g1[7] = 0;
  const i32x4 z4 = {0, 0, 0, 0};
#if __clang_major__ >= 23
  const i32x8 z8 = {0, 0, 0, 0, 0, 0, 0, 0};
  __builtin_amdgcn_tensor_load_to_lds(g0, g1, z4, z4, z8, 0);
#else
  __builtin_amdgcn_tensor_load_to_lds(g0, g1, z4, z4, 0);
#endif
}
#endif

// ---------------------------------------------------------------------------
// One bidirectional LSTM layer. grid.z = 2 (dir), one persistent workgroup
// per direction. pre = x @ wih^T + b precomputed as (B,T,4H) (gate order
// i,f,g,o). h lives in LDS (row stride 260 == 4 mod 64 -> conflict-free
// ds_load_b64 A-fragments); c lives in registers. 16 waves, each owning a
// 16-wide hidden slice, do 4 gate WMMAs (i,f,g,o) per K-step:
//   g = pre[t] + h @ whh^T   (M=8 batch rows, A rows 8..15 held at zero)
// The pre[t] tile is DMA'd into LDS by the TDM, overlapped with the
// recurrent GEMM of the same step (issue after epilogue(t-1), wait after
// gemm(t)), pulling its fetch latency off the 512-step serial chain.
// out is (B,T,2H) with fwd in [0,H), bwd in [H,2H), aligned at position t.
// ---------------------------------------------------------------------------
#define HST 260   // LDS row stride in floats (260 % 64 == 4)

struct LstmSmem {
  float hs[16 * HST];      // 16640 bytes, at LDS offset 0
  float pre[8 * 1024];     // 32768 bytes: staged (B=8, 4H=1024) tile
};
#define PRE_LDS_OFF (16 * HST * 4)

__global__ __launch_bounds__(512)
void lstm_layer(const float* __restrict__ pre_f, const float* __restrict__ pre_b,
                const float* __restrict__ whh_f, const float* __restrict__ whh_b,
                float* __restrict__ out, int T)
{
  __shared__ LstmSmem sm;
  const int dir = blockIdx.z;
  const float* __restrict__ pre = dir ? pre_b : pre_f;
  const float* __restrict__ whh = dir ? whh_b : whh_f;
  const int lane = threadIdx.x & 31;
  const int wid  = threadIdx.x >> 5;
  const int half = lane >> 4, r = lane & 15;
  const int n0 = wid << 4;  // hidden slice [n0, n0+16)

  for (int i = threadIdx.x; i < 16 * HST; i += blockDim.x) sm.hs[i] = 0.0f;
  const v8f vzero = {0,0,0,0,0,0,0,0};
  v8f c = vzero;
  __syncthreads();

#if HAVE_TDM
  if (wid == 0) {  // wave-level DMA: one wave issues for the workgroup
    const int t0 = dir ? (T - 1) : 0;
    tdm_load_pre_tile(pre + t0 * 1024, PRE_LDS_OFF, T);
  }
#endif

  const float* w0 = whh + (0 * 256 + n0 + r) * 256 + 2 * half;  // gate i
  const float* w1 = whh + (1 * 256 + n0 + r) * 256 + 2 * half;  // gate f
  const float* w2 = whh + (2 * 256 + n0 + r) * 256 + 2 * half;  // gate g
  const float* w3 = whh + (3 * 256 + n0 + r) * 256 + 2 * half;  // gate o

  for (int t = 0; t < T; t++) {
    const int tt = dir ? (T - 1 - t) : t;
    v8f g0 = vzero, g1 = vzero, g2 = vzero, g3 = vzero;
    for (int k = 0; k < 256; k += 4) {
      v2f a  = *(const v2f*)&sm.hs[r * HST + 2 * half + k];
      v2f b0 = *(const v2f*)(w0 + k);
      v2f b1 = *(const v2f*)(w1 + k);
      v2f b2 = *(const v2f*)(w2 + k);
      v2f b3 = *(const v2f*)(w3 + k);
      g0 = __builtin_amdgcn_wmma_f32_16x16x4_f32(false, a, false, b0, (short)0, g0, false, false);
      g1 = __builtin_amdgcn_wmma_f32_16x16x4_f32(false, a, false, b1, (short)0, g1, false, false);
      g2 = __builtin_amdgcn_wmma_f32_16x16x4_f32(false, a, false, b2, (short)0, g2, false, false);
      g3 = __builtin_amdgcn_wmma_f32_16x16x4_f32(false, a, false, b3, (short)0, g3, false, false);
    }
    __syncthreads();  // all waves done reading hs (and done with sm.pre)
#if HAVE_TDM
    if (wid == 0) __builtin_amdgcn_s_wait_tensorcnt(0);  // pre[t] tile landed
    __syncthreads();                                     // visible to all waves
#endif
    if (half == 0) {  // lanes 0..15 hold batch rows 0..7 in acc VGPRs 0..7
      const int n = n0 + r;
#pragma unroll
      for (int v = 0; v < 8; v++) {
#if HAVE_TDM
        const float pi = sm.pre[v * 1024 + n];
        const float pf = sm.pre[v * 1024 + 256 + n];
        const float pg = sm.pre[v * 1024 + 512 + n];
        const float po = sm.pre[v * 1024 + 768 + n];
#else
        const int base = (v * T + tt) * 1024 + n;
        const float pi = pre[base];
        const float pf = pre[base + 256];
        const float pg = pre[base + 512];
        const float po = pre[base + 768];
#endif
        const float iv = g0[v] + pi;
        const float fv = g1[v] + pf;
        const float gv = g2[v] + pg;
        const float ov = g3[v] + po;
        const float cn = sigm(fv) * c[v] + sigm(iv) * tanhf(gv);
        c[v] = cn;
        const float hv = sigm(ov) * tanhf(cn);
        sm.hs[v * HST + n] = hv;
        out[(v * T + tt) * 512 + dir * 256 + n] = hv;
      }
    }
    __syncthreads();  // hs updated + pre reads done before next step
#if HAVE_TDM
    if (wid == 0 && (t + 1) < T) {  // overlap DMA of pre[t+1] with gemm(t+1)
      const int tn = dir ? (T - 2 - t) : (t + 1);
      tdm_load_pre_tile(pre + tn * 1024, PRE_LDS_OFF, T);
    }
#endif
  }
}

// ---------------------------------------------------------------------------
// Host orchestration
// ---------------------------------------------------------------------------
extern "C" void kernel_launch(void* const* d_in, const int* in_sizes, int n_in,
                              void* d_out, int out_size, void* d_ws, size_t ws_size,
                              hipStream_t stream)
{
  (void)in_sizes; (void)n_in; (void)out_size; (void)ws_size;
  const float* nodef = (const float*)d_in[0];   // (8,512,128)
  const float* adj   = (const float*)d_in[1];   // (8,512,512)
  const float* mask  = (const float*)d_in[2];   // (8,512)

  // Workspace layout (floats). watt aliases pre_f (GNN phase vs LSTM phase);
  // MLP hidden aliases hp (proj buffer dead after GNN).
  float* ws    = (float*)d_ws;
  float* si    = ws;                           // 4096
  float* sj    = si + 4096;                    // 4096
  float* hp    = sj + 4096;                    // 4096*256  proj / MLP hidden
  float* hA    = hp + 4096 * 256;              // 4096*256
  float* hB    = hA + 4096 * 256;              // 4096*256
  float* pre_f = hB + 4096 * 256;              // 4096*1024 (aliases watt)
  float* pre_b = pre_f + 4096 * 1024;          // 4096*1024
  float* x1    = pre_b + 4096 * 1024;          // 4096*512
  float* x2    = x1 + 4096 * 512;              // 4096*512
  float* watt  = pre_f;                        // 8*512*512 (8 MB < 16 MB)
  float* hid   = hp;

  auto gemm = [&](const float* A, int lda, int sA, const float* B, int ldb, int sB,
                  const float* bias, float* C, int ldc, int sC,
                  const float* rmask, int mstride,
                  int M, int N, int K, bool bt, bool relu, int batches) {
    const int tiles = (M >> 4) * ((N + 63) >> 6);
    dim3 grid((tiles + 7) / 8, 1, batches);
    dim3 blk(256);
    if (bt) {
      if (relu) gemm_wmma<true, true><<<grid, blk, 0, stream>>>(A, lda, sA, B, ldb, sB, bias, C, ldc, sC, rmask, mstride, M, N, K);
      else      gemm_wmma<true, false><<<grid, blk, 0, stream>>>(A, lda, sA, B, ldb, sB, bias, C, ldc, sC, rmask, mstride, M, N, K);
    } else {
      if (relu) gemm_wmma<false, true><<<grid, blk, 0, stream>>>(A, lda, sA, B, ldb, sB, bias, C, ldc, sC, rmask, mstride, M, N, K);
      else      gemm_wmma<false, false><<<grid, blk, 0, stream>>>(A, lda, sA, B, ldb, sB, bias, C, ldc, sC, rmask, mstride, M, N, K);
    }
  };

  // ---- 3 GNN layers ----
  const float* xin = nodef;
  int Kin = 128;
  float* houts[3] = {hA, hB, hA};
  for (int l = 0; l < 3; l++) {
    const float* gw = (const float*)d_in[4 + 4 * l];  // (256, Kin)
    const float* gb = (const float*)d_in[5 + 4 * l];  // (256,)
    const float* aw = (const float*)d_in[6 + 4 * l];  // (1, 512)
    const float* ab = (const float*)d_in[7 + 4 * l];  // (1,)
    // hp = xin @ gw^T + gb
    gemm(xin, Kin, 0, gw, Kin, 0, gb, hp, 256, 0, nullptr, 0, 4096, 256, Kin, true, false, 1);
    attn_vec<<<dim3(16), dim3(256), 0, stream>>>(hp, aw, ab, si, sj);
    attn_watt<<<dim3(512), dim3(256), 0, stream>>>(si, sj, adj, mask, watt);
    // houts[l][b] = watt[b] @ hp[b], row-masked, ReLU for layers 0,1
    gemm(watt, 512, 512 * 512, hp, 256, 512 * 256, nullptr, houts[l], 256, 512 * 256,
         mask, 512, 512, 256, 512, false, (l < 2), 8);
    xin = houts[l];
    Kin = 256;
  }
  const float* h3 = hA;

  // ---- LSTM layer 0 (in=256) ----
  gemm(h3, 256, 0, (const float*)d_in[16], 256, 0, (const float*)d_in[18],
       pre_f, 1024, 0, nullptr, 0, 4096, 1024, 256, true, false, 1);
  gemm(h3, 256, 0, (const float*)d_in[19], 256, 0, (const float*)d_in[21],
       pre_b, 1024, 0, nullptr, 0, 4096, 1024, 256, true, false, 1);
  lstm_layer<<<dim3(1, 1, 2), dim3(512), 0, stream>>>(
      pre_f, pre_b, (const float*)d_in[17], (const float*)d_in[20], x1, 512);

  // ---- LSTM layer 1 (in=512) ----
  gemm(x1, 512, 0, (const float*)d_in[22], 512, 0, (const float*)d_in[24],
       pre_f, 1024, 0, nullptr, 0, 4096, 1024, 512, true, false, 1);
  gemm(x1, 512, 0, (const float*)d_in[25], 512, 0, (const float*)d_in[27],
       pre_b, 1024, 0, nullptr, 0, 4096, 1024, 512, true, false, 1);
  lstm_layer<<<dim3(1, 1, 2), dim3(512), 0, stream>>>(
      pre_f, pre_b, (const float*)d_in[23], (const float*)d_in[26], x2, 512);

  // ---- MLP heads (outputs concat in return order) ----
  float* out = (float*)d_out;
  struct Head { int wi; const float* x; int ind; int outd; float* o; };
  Head heads[4] = {
    {28, x2, 512, 64,  out},                 // op_pred
    {32, x2, 512, 16,  out + 4096 * 64},     // param_pred
    {36, x2, 512, 128, out + 4096 * 80},     // sketch_pred
    {40, h3, 256, 128, out + 4096 * 208},    // node_pred
  };
  for (int hI = 0; hI < 4; hI++) {
    const Head& hh = heads[hI];
    gemm(hh.x, hh.ind, 0, (const float*)d_in[hh.wi], hh.ind, 0,
         (const float*)d_in[hh.wi + 1], hid, 256, 0, nullptr, 0,
         4096, 256, hh.ind, true, true, 1);
    gemm(hid, 256, 0, (const float*)d_in[hh.wi + 2], 256, 0,
         (const float*)d_in[hh.wi + 3], hh.o, hh.outd, 0, nullptr, 0,
         4096, hh.outd, 256, true, false, 1);
  }
}